// True_nn_vib_analysis_11690900980362
// MI455X (gfx1250) — compile-verified
//
#include <hip/hip_runtime.h>
#include <math.h>

// ---------------- problem constants ----------------
#define N_ATOMS 1500
#define DEG     32
#define N_EDGES (N_ATOMS*DEG)         // 48000
#define N3      (3*N_ATOMS)           // 4500
#define NP      4512                  // padded: 16 * 282
#define NBLK    (NP/16)               // 282 blocks of 16
#define NPAIRS  (NBLK/2)              // 141 disjoint block pairs / round
#define NROUNDS (NBLK-1)              // 281 tournament rounds / sweep
#define NSWEEPS 8
#define NMODES  (N3-6)                // 4494
#define PAD_EIG 1.0e30f

#define HESS_T   9.375829e+28f
#define CM_HZ    2.99792458e10f
#define IR_COFF  42.2561f
#define RAMAN_COFF 0.078424f
#define SCALE_F  0.965f

// row-update kernel: grid.x workgroups * 8 waves stride over the 282 col tiles
#define ROWWG    6
#define RSTRIDE  (ROWWG*8)            // 48

typedef float v2f __attribute__((ext_vector_type(2)));
typedef float v8f __attribute__((ext_vector_type(8)));

__constant__ float c_masses[10] = {
    1.008f, 4.003f, 6.941f, 9.012f, 10.811f,
    12.011f, 14.007f, 15.999f, 18.998f, 20.18f };

// Round-robin ("tournament") pairing: m even teams, m-1 rounds, m/2 disjoint
// pairs per round.
__device__ __forceinline__ void pair_blocks(int round, int k, int m, int& p, int& q) {
    int a, b;
    if (k == 0) { a = m - 1; b = round % (m - 1); }
    else {
        a = (round + k) % (m - 1);
        b = (round - k + (m - 1)) % (m - 1);
    }
    p = a < b ? a : b;
    q = a < b ? b : a;
}

// generic->LDS offset (low 32 bits of a shared-memory generic pointer are the
// in-wave LDS byte offset per the aperture rules)
__device__ __forceinline__ unsigned lds_off32(const void* p) {
    return (unsigned)(size_t)p;
}

// ---------------- Hessian assembly ----------------
__global__ __launch_bounds__(256) void init_kernel(float* __restrict__ H, float* __restrict__ V) {
    int idx = blockIdx.x * 256 + threadIdx.x;
    if (idx >= NP * NP) return;
    int r = idx / NP, c = idx % NP;
    H[idx] = 0.0f;
    V[idx] = (r == c) ? 1.0f : 0.0f;
}

__global__ __launch_bounds__(256) void scatter_hij_kernel(const long long* __restrict__ ei,
                                                          const float* __restrict__ Hij,
                                                          float* __restrict__ H) {
    int idx = blockIdx.x * 256 + threadIdx.x;
    if (idx >= N_EDGES * 9) return;
    int e = idx / 9, ab = idx % 9, a = ab / 3, b = ab % 3;
    int i = (int)ei[e];              // edge_index[0][e]
    int j = (int)ei[N_EDGES + e];    // edge_index[1][e]
    H[(size_t)(3 * j + a) * NP + (3 * i + b)] = Hij[idx];
}

__global__ __launch_bounds__(256) void diag_hi_kernel(const float* __restrict__ Hi,
                                                      float* __restrict__ H) {
    int idx = blockIdx.x * 256 + threadIdx.x;
    if (idx >= N_ATOMS * 9) return;
    int n = idx / 9, ab = idx % 9, a = ab / 3, b = ab % 3;
    H[(size_t)(3 * n + a) * NP + (3 * n + b)] = Hi[idx];
}

__global__ __launch_bounds__(256) void wm_kernel(const int* __restrict__ z,
                                                 float* __restrict__ wm) {
    int k = blockIdx.x * 256 + threadIdx.x;
    if (k >= NP) return;
    if (k < N3) {
        int n = k / 3;
        wm[k] = rsqrtf(c_masses[z[n]]);
    } else {
        wm[k] = 1.0f;
    }
}

// H <- 0.5*(H + H^T) * wm wm^T ; pad diagonal pushed to +huge so the 12 pad
// eigenmodes sort past all physical ones.
__global__ __launch_bounds__(256) void symw_kernel(float* __restrict__ H,
                                                   const float* __restrict__ wm) {
    int idx = blockIdx.x * 256 + threadIdx.x;
    if (idx >= NP * NP) return;
    int r = idx / NP, c = idx % NP;
    if (r > c) return;
    if (r == c && r >= N3) { H[(size_t)r * NP + r] = PAD_EIG; return; }
    float s = 0.5f * (H[(size_t)r * NP + c] + H[(size_t)c * NP + r]) * wm[r] * wm[c];
    H[(size_t)r * NP + c] = s;
    H[(size_t)c * NP + r] = s;
}

// ---------------- blocked two-sided Jacobi ----------------
// Stage 1: per block-pair (p,q), gather the 32x32 pivot block, diagonalize it
// with cyclic scalar Jacobi in LDS, emit U *transposed* (Ut[c*32+k] = U[k][c])
// so the GEMM kernels can fetch WMMA operands as contiguous b64 LDS loads.
__global__ __launch_bounds__(256) void jacobi_rot_kernel(const float* __restrict__ H,
                                                         float* __restrict__ Uws,
                                                         int round) {
    __shared__ float S[32][33];
    __shared__ float Um[32][33];
    __shared__ float cc[16], ss[16];
    __shared__ int   pi_[16], pj_[16];

    int pair = blockIdx.x;
    int p, q; pair_blocks(round, pair, NBLK, p, q);
    int t = threadIdx.x;

    for (int idx = t; idx < 32 * 32; idx += 256) {
        int r = idx >> 5, c = idx & 31;
        int gr = (r < 16) ? (p * 16 + r) : (q * 16 + (r - 16));
        int gc = (c < 16) ? (p * 16 + c) : (q * 16 + (c - 16));
        S[r][c]  = H[(size_t)gr * NP + gc];
        Um[r][c] = (r == c) ? 1.0f : 0.0f;
    }
    __syncthreads();

    for (int sweep = 0; sweep < 6; ++sweep) {
        for (int rr = 0; rr < 31; ++rr) {
            if (t < 16) {
                int i, j; pair_blocks(rr, t, 32, i, j);
                float app = S[i][i], aqq = S[j][j], apq = S[i][j];
                float cr = 1.0f, sr = 0.0f;
                if (fabsf(apq) > 1e-28f) {
                    float tau = (aqq - app) / (2.0f * apq);
                    float tt = (tau >= 0.0f ? 1.0f : -1.0f)
                               / (fabsf(tau) + sqrtf(1.0f + tau * tau));
                    cr = rsqrtf(1.0f + tt * tt);
                    sr = tt * cr;
                }
                cc[t] = cr; ss[t] = sr; pi_[t] = i; pj_[t] = j;
            }
            __syncthreads();
            // S <- J^T S  (disjoint row pairs -> race free)
            for (int w = t; w < 512; w += 256) {
                int pr = w >> 5, col = w & 31;
                int i = pi_[pr], j = pj_[pr];
                float cr = cc[pr], sr = ss[pr];
                float a = S[i][col], b = S[j][col];
                S[i][col] = cr * a - sr * b;
                S[j][col] = sr * a + cr * b;
            }
            __syncthreads();
            // S <- S J ; U <- U J  (disjoint column pairs)
            for (int w = t; w < 512; w += 256) {
                int pr = w >> 5, row = w & 31;
                int i = pi_[pr], j = pj_[pr];
                float cr = cc[pr], sr = ss[pr];
                float a = S[row][i], b = S[row][j];
                S[row][i] = cr * a - sr * b;
                S[row][j] = sr * a + cr * b;
                float ua = Um[row][i], ub = Um[row][j];
                Um[row][i] = cr * ua - sr * ub;
                Um[row][j] = sr * ua + cr * ub;
            }
            __syncthreads();
        }
    }
    float* Ut = Uws + pair * 1024;
    for (int idx = t; idx < 1024; idx += 256)        // Ut[c*32+k] = U[k][c]
        Ut[idx] = Um[idx & 31][idx >> 5];
}

// Stage 2: block-row update  A[rows(p,q), :] <- U^T * A[rows(p,q), :]
// 8-wave workgroups: U^T loaded to LDS once, each wave strides over column
// tiles with a double-buffered A tile staged by global_load_async_to_lds_b32
// (ASYNCcnt pipelining: next tile streams from L2 while WMMAs run).
__global__ __launch_bounds__(256) void jacobi_row_kernel(float* __restrict__ H,
                                                         const float* __restrict__ Uws,
                                                         int round) {
    __shared__ float sUt[32][36];            // sUt[c][k] = U[k][c]; 144B rows
    __shared__ float sAt[8][2][16][36];      // per-wave double buffer, [c][r]
    int pair = blockIdx.y;
    int p, q; pair_blocks(round, pair, NBLK, p, q);
    const float* Ut = Uws + pair * 1024;

    __builtin_prefetch(Ut, 0, 3);            // global_prefetch_b8

    {   // cooperative Ut -> LDS copy: one b128 per thread
        int t4 = threadIdx.x * 4;
        *(float4*)&sUt[t4 >> 5][t4 & 31] = *(const float4*)(Ut + t4);
    }
    __syncthreads();

    int wave = threadIdx.x >> 5;
    int lane = threadIdx.x & 31;
    int m = lane & 15, h = lane >> 4;
    int wg = blockIdx.x * 8 + wave;          // 0..47, strides the 282 tiles

    unsigned lds0 = lds_off32(&sAt[wave][0][0][0]);
    unsigned lds1 = lds_off32(&sAt[wave][1][0][0]);

    // async-stage the 32x16 tile at column-tile ct, transposed, into ldsb
    auto stage = [&](int ct, unsigned ldsb) {
#pragma unroll
        for (int e = 0; e < 16; ++e) {
            int idx = e * 32 + lane;
            int r = idx >> 4, c = idx & 15;
            int gr = (r < 16) ? (p * 16 + r) : (q * 16 + (r - 16));
            const float* gp = H + (size_t)gr * NP + ct * 16 + c;
            unsigned lo = ldsb + (unsigned)((c * 36 + r) * 4);
            asm volatile("global_load_async_to_lds_b32 %0, %1, off"
                         :: "v"(lo), "v"(gp) : "memory");
        }
    };

    stage(wg, lds0);
    for (int ct = wg, bi = 0; ct < NBLK; ct += RSTRIDE, bi ^= 1) {
        int ctn = ct + RSTRIDE;
        if (ctn < NBLK) {
            stage(ctn, bi ? lds0 : lds1);
            asm volatile("s_wait_asynccnt 0x10" ::: "memory"); // tile ct done
        } else {
            asm volatile("s_wait_asynccnt 0x0" ::: "memory");
        }
        const float (*At)[36] = sAt[wave][bi];

        v8f acc0 = {0, 0, 0, 0, 0, 0, 0, 0};
        v8f acc1 = {0, 0, 0, 0, 0, 0, 0, 0};
        // out0[m][n] = sum_k U[k][m]   * A[k][n]   (rows -> block p)
        // out1[m][n] = sum_k U[k][m+16]* A[k][n]   (rows -> block q)
#pragma unroll
        for (int k0 = 0; k0 < 32; k0 += 4) {
            int ka = k0 + 2 * h;                       // even -> 8B aligned
            v2f b  = *(const v2f*)&At[m][ka];          // A[ka..ka+1][n]
            v2f a0 = *(const v2f*)&sUt[m][ka];         // U[ka..ka+1][m]
            v2f a1 = *(const v2f*)&sUt[16 + m][ka];
            acc0 = __builtin_amdgcn_wmma_f32_16x16x4_f32(false, a0, false, b,
                                                         (short)0, acc0, false, false);
            acc1 = __builtin_amdgcn_wmma_f32_16x16x4_f32(false, a1, false, b,
                                                         (short)0, acc1, false, false);
        }
        int c0 = ct * 16;
#pragma unroll
        for (int r = 0; r < 8; ++r) {
            int M = r + 8 * h;
            H[(size_t)(p * 16 + M) * NP + c0 + m] = acc0[r];
            H[(size_t)(q * 16 + M) * NP + c0 + m] = acc1[r];
        }
    }
}

// Stage 3: block-col update  M[:, cols(p,q)] <- M[:, cols(p,q)] * U
// 8-wave workgroups share one U^T in LDS; blockIdx.z selects H (0) / V (1).
__global__ __launch_bounds__(256) void jacobi_col_kernel(float* __restrict__ Hm,
                                                         float* __restrict__ Vm,
                                                         const float* __restrict__ Uws,
                                                         int round) {
    __shared__ float sUt[32][36];            // sUt[c][k] = U[k][c]
    __shared__ float sA[8][16][36];          // per-wave row-major 16x32 tile
    float* Mtx = blockIdx.z ? Vm : Hm;
    int pair = blockIdx.y;
    int p, q; pair_blocks(round, pair, NBLK, p, q);
    const float* Ut = Uws + pair * 1024;

    {
        int t4 = threadIdx.x * 4;
        *(float4*)&sUt[t4 >> 5][t4 & 31] = *(const float4*)(Ut + t4);
    }
    __syncthreads();

    int wave = threadIdx.x >> 5;
    int lane = threadIdx.x & 31;
    int rt = blockIdx.x * 8 + wave;
    if (rt >= NBLK) return;
    int r0 = rt * 16;

    // stage the 16x32 tile (rows r0.., pair columns): 4x b128 per lane
#pragma unroll
    for (int e = 0; e < 4; ++e) {
        int idx = e * 32 + lane;             // float4 index 0..127
        int r = idx >> 3, c4 = (idx & 7) * 4;
        int gc = (c4 < 16) ? (p * 16 + c4) : (q * 16 + (c4 - 16));
        *(float4*)&sA[wave][r][c4] = *(const float4*)&Mtx[(size_t)(r0 + r) * NP + gc];
    }

    int m = lane & 15, h = lane >> 4;
    v8f acc0 = {0, 0, 0, 0, 0, 0, 0, 0};
    v8f acc1 = {0, 0, 0, 0, 0, 0, 0, 0};
    // out0[r][n] = sum_k A[r][k] * U[k][n]     (cols -> block p)
    // out1[r][n] = sum_k A[r][k] * U[k][n+16]  (cols -> block q)
#pragma unroll
    for (int k0 = 0; k0 < 32; k0 += 4) {
        int ka = k0 + 2 * h;
        v2f a  = *(const v2f*)&sA[wave][m][ka];        // A[m][ka..ka+1]
        v2f b0 = *(const v2f*)&sUt[m][ka];             // U[ka..ka+1][n]
        v2f b1 = *(const v2f*)&sUt[16 + m][ka];
        acc0 = __builtin_amdgcn_wmma_f32_16x16x4_f32(false, a, false, b0,
                                                     (short)0, acc0, false, false);
        acc1 = __builtin_amdgcn_wmma_f32_16x16x4_f32(false, a, false, b1,
                                                     (short)0, acc1, false, false);
    }
#pragma unroll
    for (int r = 0; r < 8; ++r) {
        int M = r + 8 * h;
        Mtx[(size_t)(r0 + M) * NP + p * 16 + m] = acc0[r];
        Mtx[(size_t)(r0 + M) * NP + q * 16 + m] = acc1[r];
    }
}

// ---------------- spectrum extraction ----------------
__global__ __launch_bounds__(256) void diag_extract_kernel(const float* __restrict__ H,
                                                           float* __restrict__ eva) {
    int i = blockIdx.x * 256 + threadIdx.x;
    if (i >= NP) return;
    eva[i] = H[(size_t)i * NP + i];
}

// rank[i] = position of eigenvalue i in ascending order (tie-broken by index)
__global__ __launch_bounds__(256) void rank_kernel(const float* __restrict__ eva,
                                                   int* __restrict__ rank) {
    __shared__ int red[256];
    int i = blockIdx.x;
    float ei = eva[i];
    int cnt = 0;
    for (int j = threadIdx.x; j < NP; j += 256) {
        float ej = eva[j];
        if (ej < ei || (ej == ei && j < i)) cnt++;
    }
    red[threadIdx.x] = cnt;
    __syncthreads();
    for (int s = 128; s > 0; s >>= 1) {
        if (threadIdx.x < s) red[threadIdx.x] += red[threadIdx.x + s];
        __syncthreads();
    }
    if (threadIdx.x == 0) rank[i] = red[0];
}

// One workgroup per eigenpair: mode = -V[:,i] * wm, project onto dedipole
// (3 dipole derivs) and depolar (6 polar derivs), reduce, emit freq/ir/raman.
__global__ __launch_bounds__(256) void spectra_kernel(const float* __restrict__ V,
                                                      const float* __restrict__ wm,
                                                      const float* __restrict__ eva,
                                                      const int* __restrict__ rank,
                                                      const float* __restrict__ dedipole,
                                                      const float* __restrict__ depolar,
                                                      float* __restrict__ out) {
    __shared__ float red[9][257];
    int i = blockIdx.x;
    int rk = rank[i];
    if (rk < 6 || rk >= N3) return;      // skip 6 rigid modes and 12 pad modes
    int mo = rk - 6;

    float s[9];
#pragma unroll
    for (int u = 0; u < 9; ++u) s[u] = 0.0f;
    for (int k = threadIdx.x; k < N3; k += 256) {
        float pk = -V[(size_t)k * NP + i] * wm[k];
        const float* dd = dedipole + (size_t)k * 3;   // [(n*3+a)*3 + b] = k*3+b
        s[0] += pk * dd[0]; s[1] += pk * dd[1]; s[2] += pk * dd[2];
        const float* dp = depolar + (size_t)k * 6;    // k*6+b
#pragma unroll
        for (int b = 0; b < 6; ++b) s[3 + b] += pk * dp[b];
    }
#pragma unroll
    for (int u = 0; u < 9; ++u) red[u][threadIdx.x] = s[u];
    __syncthreads();
    for (int st = 128; st > 0; st >>= 1) {
        if (threadIdx.x < st)
#pragma unroll
            for (int u = 0; u < 9; ++u)
                red[u][threadIdx.x] += red[u][threadIdx.x + st];
        __syncthreads();
    }
    if (threadIdx.x == 0) {
        float freq = sqrtf(eva[i] * HESS_T)
                   * (SCALE_F / (2.0f * 3.14159265358979f * CM_HZ));
        float ir = IR_COFF * (red[0][0] * red[0][0] + red[1][0] * red[1][0]
                            + red[2][0] * red[2][0]);
        float xx = red[3][0], xy = red[4][0], yy = red[5][0];
        float xz = red[6][0], zy = red[7][0], zz = red[8][0];
        float alpha = (xx + yy + zz) * (1.0f / 3.0f);
        float gsq = 0.5f * ((xx - yy) * (xx - yy) + (yy - zz) * (yy - zz)
                          + (zz - xx) * (zz - xx))
                  + 3.0f * (xy * xy + xz * xz + zy * zy);
        float raman = (45.0f * alpha * alpha + 7.0f * gsq) * RAMAN_COFF;
        out[mo]              = freq;
        out[NMODES + mo]     = ir;
        out[2 * NMODES + mo] = raman;
    }
}

// ---------------- host launcher ----------------
extern "C" void kernel_launch(void* const* d_in, const int* in_sizes, int n_in,
                              void* d_out, int out_size, void* d_ws, size_t ws_size,
                              hipStream_t stream) {
    (void)in_sizes; (void)n_in; (void)out_size; (void)ws_size;
    // inputs: 0 pos (unused), 1 z (i32), 2 edge_index (i64), 3 Hi, 4 Hij,
    //         5 dedipole, 6 depolar
    const int*        z        = (const int*)d_in[1];
    const long long*  ei       = (const long long*)d_in[2];
    const float*      Hi       = (const float*)d_in[3];
    const float*      Hij      = (const float*)d_in[4];
    const float*      dedipole = (const float*)d_in[5];
    const float*      depolar  = (const float*)d_in[6];
    float*            out      = (float*)d_out;

    // workspace layout (~164 MB; H+V stay resident in the 192 MB L2)
    float* H   = (float*)d_ws;
    float* V   = H + (size_t)NP * NP;
    float* Uws = V + (size_t)NP * NP;
    float* wm  = Uws + (size_t)NPAIRS * 1024;
    float* eva = wm + NP;
    int*   rnk = (int*)(eva + NP);

    const int nHV = NP * NP;
    init_kernel<<<(nHV + 255) / 256, 256, 0, stream>>>(H, V);
    scatter_hij_kernel<<<(N_EDGES * 9 + 255) / 256, 256, 0, stream>>>(ei, Hij, H);
    diag_hi_kernel<<<(N_ATOMS * 9 + 255) / 256, 256, 0, stream>>>(Hi, H);
    wm_kernel<<<(NP + 255) / 256, 256, 0, stream>>>(z, wm);
    symw_kernel<<<(nHV + 255) / 256, 256, 0, stream>>>(H, wm);

    for (int sweep = 0; sweep < NSWEEPS; ++sweep) {
        for (int r = 0; r < NROUNDS; ++r) {
            jacobi_rot_kernel<<<NPAIRS, 256, 0, stream>>>(H, Uws, r);
            jacobi_row_kernel<<<dim3(ROWWG, NPAIRS), 256, 0, stream>>>(H, Uws, r);
            jacobi_col_kernel<<<dim3(36, NPAIRS, 2), 256, 0, stream>>>(H, V, Uws, r);
        }
    }

    diag_extract_kernel<<<(NP + 255) / 256, 256, 0, stream>>>(H, eva);
    rank_kernel<<<NP, 256, 0, stream>>>(eva, rnk);
    spectra_kernel<<<NP, 256, 0, stream>>>(V, wm, eva, rnk, dedipole, depolar, out);
}